// AngleLinear_84404697301345
// MI455X (gfx1250) — compile-verified
//
#include <hip/hip_runtime.h>
#include <math.h>

typedef __attribute__((ext_vector_type(2))) float v2f;
typedef __attribute__((ext_vector_type(8))) float v8f;

#define PI_F 3.141592653f
#define TILES_PER_WAVE 4   // main path: each wave does 16 rows x 64 cols

// ---------------- Kernel 1: row L2 norms of x [N,D] ----------------
__global__ void xnorm_kernel(const float* __restrict__ x, float* __restrict__ xn,
                             int N, int D) {
    int i = blockIdx.x * blockDim.x + threadIdx.x;
    if (i >= N) return;
    const float* row = x + (size_t)i * D;
    float s = 0.f;
    for (int d = 0; d < D; ++d) { float v = row[d]; s += v * v; }
    xn[i] = sqrtf(s);
}

// ---------------- Kernel 2: WMMA fp32 GEMM + fused colnorm + epilogue -------
// Block = 256 threads (8 waves). Wave w owns row-tile (blockIdx.y*8 + w) and
// NT column tiles at tileOffset + blockIdx.x*NT (launcher guarantees range).
// B addressing: 32-bit element indices off the w base pointer (weight < 4GB),
// lowered to GVS loads (saddr + scale_offset voffset); advanced 4*C per step.
// Column norms are accumulated as v2f (pairs are adjacent VGPRs -> v_pk_fma
// with no packing movs): half h=0 sums rows {0,1} mod 4, h=1 sums {2,3};
// shfl_xor(16) completes the per-column sum in the epilogue.
template <int NT>
__global__ __launch_bounds__(256)
void angle_gemm_kernel(const float* __restrict__ x, const float* __restrict__ w,
                       const float* __restrict__ xn, float* __restrict__ out,
                       int N, int D, int C, int tileOffset) {
    const int lane   = threadIdx.x & 31;
    const int wave   = threadIdx.x >> 5;
    const int half   = lane >> 4;    // 0: lanes 0-15, 1: lanes 16-31
    const int lane16 = lane & 15;

    const int r0 = (blockIdx.y * 8 + wave) * 16;
    if (r0 >= N) return;                       // wave-uniform exit

    const int colTileBase = tileOffset + blockIdx.x * NT;
    const int cBase = colTileBase << 4;

    v8f zero = {};
    v2f zero2 = {};
    v8f acc[NT];
    v2f cn2[NT];
#pragma unroll
    for (int t = 0; t < NT; ++t) { acc[t] = zero; cn2[t] = zero2; }

    // A: row (r0+lane16), K pair {k+2h, k+2h+1}; 8B-aligned (D even, k mult 4)
    const float* aPtr = x + (size_t)(r0 + lane16) * D + 2 * half;

    // B: 32-bit element indices; idx0 -> row (k+2h), idx1 -> row (k+2h+1)
    const uint32_t uC    = (uint32_t)C;
    const uint32_t stepC = 4u * uC;                       // advance 4 K-rows
    uint32_t idx0 = (uint32_t)(2 * half) * uC + (uint32_t)(cBase + lane16);
    uint32_t idx1 = idx0 + uC;

    // ---- prologue: load k = 0 operands ----
    v2f a = *(const v2f*)(aPtr);
    v2f b[NT];
#pragma unroll
    for (int t = 0; t < NT; ++t) {
        b[t].x = w[idx0 + 16u * t];
        b[t].y = w[idx1 + 16u * t];
    }
    idx0 += stepC; idx1 += stepC;

    // ---- main loop: preload k+4 while multiplying k (register double-buffer)
#pragma unroll 2
    for (int k = 0; k < D - 4; k += 4) {
        v2f a_n = *(const v2f*)(aPtr + k + 4);
        v2f b_n[NT];
#pragma unroll
        for (int t = 0; t < NT; ++t) {
            b_n[t].x = w[idx0 + 16u * t];
            b_n[t].y = w[idx1 + 16u * t];
        }
        idx0 += stepC; idx1 += stepC;
#pragma unroll
        for (int t = 0; t < NT; ++t) {
            cn2[t] = b[t] * b[t] + cn2[t];   // v_pk_fma_f32, no packing movs
            acc[t] = __builtin_amdgcn_wmma_f32_16x16x4_f32(
                false, a, false, b[t], (short)0, acc[t], false, false);
        }
        a = a_n;
#pragma unroll
        for (int t = 0; t < NT; ++t) b[t] = b_n[t];
    }
    // ---- final k step ----
#pragma unroll
    for (int t = 0; t < NT; ++t) {
        cn2[t] = b[t] * b[t] + cn2[t];
        acc[t] = __builtin_amdgcn_wmma_f32_16x16x4_f32(
            false, a, false, b[t], (short)0, acc[t], false, false);
    }

    // ---- epilogue: out[r,c] = clip(S/(xn[r]*cn[c]), -1, 1) * xn[r] ----
    float xnr[8];
#pragma unroll
    for (int v = 0; v < 8; ++v) xnr[v] = xn[r0 + v + 8 * half];

#pragma unroll
    for (int t = 0; t < NT; ++t) {
        float part = cn2[t].x + cn2[t].y;                 // this half's row sums
        float cnv = sqrtf(part + __shfl_xor(part, 16, 32)); // full column norm
        uint32_t oIdx = (uint32_t)(r0 + 8 * half) * uC + (uint32_t)(cBase + 16 * t + lane16);
#pragma unroll
        for (int v = 0; v < 8; ++v) {
            float rxn = xnr[v];
            float cosv = fminf(fmaxf(acc[t][v] / (rxn * cnv), -1.f), 1.f);
            out[oIdx] = cosv * rxn;
            oIdx += uC;
        }
    }
}

// ---------------- Kernel 3: per-row target-column margin correction --------
// One wave per row; computes the target-column dot AND its column norm.
__global__ void correction_kernel(const float* __restrict__ x, const int* __restrict__ target,
                                  const float* __restrict__ w, const float* __restrict__ xn,
                                  float* __restrict__ out,
                                  int N, int D, int C, float inv_one_plus_lambda) {
    int i = blockIdx.x;
    if (i >= N) return;
    int lane = threadIdx.x;
    int tgt = target[i];
    const float* row = x + (size_t)i * D;
    float s = 0.f, q = 0.f;
    for (int d = lane; d < D; d += 32) {
        float wv = w[(size_t)d * C + tgt];
        s += row[d] * wv;
        q += wv * wv;
    }
#pragma unroll
    for (int off = 16; off >= 1; off >>= 1) {
        s += __shfl_xor(s, off, 32);
        q += __shfl_xor(q, off, 32);
    }
    if (lane == 0) {
        float rxn = xn[i];
        float c_t = fminf(fmaxf(s / (rxn * sqrtf(q)), -1.f), 1.f);
        float c2 = c_t * c_t;
        float cos_m = 8.f * c2 * c2 - 8.f * c2 + 1.f;     // cos(4θ)
        float theta = acosf(c_t);
        float kf = floorf(4.f * theta / PI_F);
        float sign = 1.f - 2.f * fmodf(kf, 2.f);           // (-1)^k
        float phi = sign * cos_m - 2.f * kf;
        out[(size_t)i * C + tgt] += (phi - c_t) * rxn * inv_one_plus_lambda;
    }
}

extern "C" void kernel_launch(void* const* d_in, const int* in_sizes, int n_in,
                              void* d_out, int out_size, void* d_ws, size_t ws_size,
                              hipStream_t stream) {
    const float* x      = (const float*)d_in[0];   // [N, D]
    const int*   target = (const int*)d_in[1];     // [N]
    const float* weight = (const float*)d_in[2];   // [D, C]
    float* out = (float*)d_out;                    // [N, C]

    int N = in_sizes[1];
    int D = in_sizes[0] / N;
    int C = in_sizes[2] / D;

    float* xn = (float*)d_ws;                      // [N] floats of scratch

    xnorm_kernel<<<(N + 255) / 256, 256, 0, stream>>>(x, xn, N, D);

    int nColTiles  = C / 16;                       // C = 100000 -> 6250 tiles
    int fullBlocks = nColTiles / TILES_PER_WAVE;   // 1562 blocks of 4 tiles
    int remTiles   = nColTiles % TILES_PER_WAVE;   // 2 leftover tiles
    int rowBlocks  = (N + 127) / 128;

    if (fullBlocks) {
        dim3 grid(fullBlocks, rowBlocks);
        angle_gemm_kernel<TILES_PER_WAVE><<<grid, 256, 0, stream>>>(
            x, weight, xn, out, N, D, C, 0);
    }
    if (remTiles) {
        dim3 grid(remTiles, rowBlocks);
        angle_gemm_kernel<1><<<grid, 256, 0, stream>>>(
            x, weight, xn, out, N, D, C, fullBlocks * TILES_PER_WAVE);
    }

    // lambda = max(5, 1500/(1+0.1*1)) = 1363.6363...
    float invl = 1.0f / (1.0f + fmaxf(5.0f, 1500.0f / 1.1f));
    correction_kernel<<<N, 32, 0, stream>>>(x, target, weight, xn, out, N, D, C, invl);
}